// EFELoss_68693706932231
// MI455X (gfx1250) — compile-verified
//
#include <hip/hip_runtime.h>

#define T_    2048
#define H_    30
#define W_    30
#define C_    10
#define CELLS (H_ * W_)
#define TOTAL (T_ * CELLS)
#define NBLK  ((TOTAL + 255) / 256)   // 7200 main-pass blocks
#define EPS_    1e-8f
#define NEG_    (-1e9f)
#define THRESH_ 0.7f

typedef float v2f __attribute__((ext_vector_type(2)));
typedef float v8f __attribute__((ext_vector_type(8)));

// ---- workspace layout (float offsets into d_ws) ----
// [0..15]                  logc (10 used)
// [32..931]                pos_conf sum per cell
// [1024 .. 1024+NBLK)      per-block kl partials
// [8704 .. 8704+NBLK)      per-block bi partials
// [16384 .. 16384+TOTAL)   per-element fprobs.max scratch
// bytes after that:        colors (uint8, TOTAL bytes)
#define WS_LOGC    0
#define WS_POSCONF 32
#define WS_PART_KL 1024
#define WS_PART_BI 8704
#define WS_FMAX    16384

// ---------------------------------------------------------------------------
// k0: pref_logits = pe @ W_pref + b_pref  via V_WMMA_F32_16X16X4_F32 (1 wave),
//     then logc = log_softmax(pref_logits).
// ---------------------------------------------------------------------------
__global__ __launch_bounds__(32)
void k0_pref(const float* __restrict__ pe, const float* __restrict__ Wp,
             const float* __restrict__ bpref, float* __restrict__ ws) {
  const int lane = threadIdx.x;          // exactly 32 threads = 1 wave
  v8f acc = {};
  const int n  = lane & 15;              // output column this lane owns
  const int kb = (lane < 16) ? 0 : 2;    // K sub-offset per ISA A/B layout
  for (int k0 = 0; k0 < 256; k0 += 4) {
    v2f a = {0.f, 0.f};
    v2f b = {0.f, 0.f};
    // A is 16x4, only row M=0 nonzero (= pe slice). Lane0 holds K=0,1; lane16 K=2,3.
    if (lane == 0)  { a.x = pe[k0 + 0]; a.y = pe[k0 + 1]; }
    if (lane == 16) { a.x = pe[k0 + 2]; a.y = pe[k0 + 3]; }
    // B is 4x16: lane n holds column n; VGPR0 = K=kb, VGPR1 = K=kb+1.
    if (n < C_) {
      b.x = Wp[(k0 + kb + 0) * C_ + n];
      b.y = Wp[(k0 + kb + 1) * C_ + n];
    }
    acc = __builtin_amdgcn_wmma_f32_16x16x4_f32(false, a, false, b,
                                                (short)0, acc, false, false);
  }
  // D row M=0 lives in acc[0] on lanes 0..15.
  __shared__ float sl[16];
  if (lane < 16) sl[lane] = (lane < C_) ? (acc[0] + bpref[lane]) : 0.f;
  __syncthreads();
  if (lane < C_) {
    float m = sl[0];
    #pragma unroll
    for (int c = 1; c < C_; ++c) m = fmaxf(m, sl[c]);
    float s = 0.f;
    #pragma unroll
    for (int c = 0; c < C_; ++c) s += __expf(sl[c] - m);
    ws[WS_LOGC + lane] = sl[lane] - m - __logf(s);
  }
}

// ---------------------------------------------------------------------------
// k1: fused main pass over (t, cell). Writes conf, colors, fmax, and one
//     deterministic (kl, bi) partial per block.
// ---------------------------------------------------------------------------
__global__ __launch_bounds__(256)
void k1_main(const float* __restrict__ fpred, const float* __restrict__ bpred,
             const int* __restrict__ gmask, float* __restrict__ ws,
             unsigned char* __restrict__ colors, float* __restrict__ out_conf) {
  const int tid = threadIdx.x;
  const int i   = blockIdx.x * 256 + tid;
  float kl_p = 0.f, bi_p = 0.f;

  if (i < TOTAL) {
    const int  cell = i % CELLS;
    const bool mk   = gmask[cell] > 0;
    const float* fr = fpred + (size_t)i * C_;
    const float* br = bpred + (size_t)i * C_;

    float f[C_], b[C_];
    #pragma unroll
    for (int c = 0; c < C_; ++c) {
      f[c] = mk ? fr[c] : NEG_;
      b[c] = mk ? br[c] : NEG_;
    }
    float mf = f[0], mb = b[0];
    #pragma unroll
    for (int c = 1; c < C_; ++c) { mf = fmaxf(mf, f[c]); mb = fmaxf(mb, b[c]); }
    float ef[C_], eb[C_], sf = 0.f, sb = 0.f;
    #pragma unroll
    for (int c = 0; c < C_; ++c) {
      ef[c] = __expf(f[c] - mf); sf += ef[c];
      eb[c] = __expf(b[c] - mb); sb += eb[c];
    }
    const float isf = 1.f / sf, isb = 1.f / sb;
    const float lsf = __logf(sf);

    float Hf = 0.f, Hb = 0.f, js = 0.f, kl = 0.f, fmax = 0.f;
    float best = f[0];
    int   amax = 0;
    #pragma unroll
    for (int c = 0; c < C_; ++c) {
      const float pf = ef[c] * isf;
      const float pb = eb[c] * isb;
      const float lpf = __logf(pf + EPS_);
      const float lpb = __logf(pb + EPS_);
      const float lmid = __logf(0.5f * (pf + pb) + EPS_);
      Hf -= pf * lpf;
      Hb -= pb * lpb;
      js += 0.5f * pf * (lpf - lmid) + 0.5f * pb * (lpb - lmid);
      fmax = fmaxf(fmax, pf);
      if (f[c] > best) { best = f[c]; amax = c; }
      kl += pf * ((f[c] - mf - lsf) - ws[WS_LOGC + c]);
    }
    const float conf = 1.f / (1.f + 0.5f * (Hf + Hb));
    out_conf[i]      = conf;
    ws[WS_FMAX + i]  = fmax;
    colors[i]        = (unsigned char)amax;
    if (mk) { kl_p = kl; bi_p = js * conf; }
  }

  __shared__ float skl[256], sbi[256];
  skl[tid] = kl_p; sbi[tid] = bi_p;
  __syncthreads();
  #pragma unroll
  for (int s = 128; s > 0; s >>= 1) {
    if (tid < s) { skl[tid] += skl[tid + s]; sbi[tid] += sbi[tid + s]; }
    __syncthreads();
  }
  if (tid == 0) {
    ws[WS_PART_KL + blockIdx.x] = skl[0];   // deterministic per-block slot
    ws[WS_PART_BI + blockIdx.x] = sbi[0];
  }
}

// ---------------------------------------------------------------------------
// k2: movement field from argmax colors.
// ---------------------------------------------------------------------------
__global__ __launch_bounds__(256)
void k2_move(const unsigned char* __restrict__ colors, float* __restrict__ out_mv) {
  const int i = blockIdx.x * 256 + threadIdx.x;
  if (i >= TOTAL) return;
  const int t = i / CELLS, cell = i % CELLS;
  const int y = cell / W_, x = cell % W_;
  float my = 0.f, mx = 0.f;
  if (t < T_ - 1 && y > 0 && y < H_ - 1 && x > 0 && x < W_ - 1) {
    const int cur = colors[i];
    const unsigned char* nxt = colors + (size_t)(t + 1) * CELLS;
    float cnt = 0.f, sdy = 0.f, sdx = 0.f;
    #pragma unroll
    for (int dy = -1; dy <= 1; ++dy) {
      #pragma unroll
      for (int dx = -1; dx <= 1; ++dx) {
        if ((int)nxt[(y + dy) * W_ + (x + dx)] == cur) {
          cnt += 1.f; sdy += (float)dy; sdx += (float)dx;
        }
      }
    }
    if (cnt > 0.f) { my = sdy / cnt; mx = sdx / cnt; }
  }
  out_mv[(size_t)2 * i + 0] = my;
  out_mv[(size_t)2 * i + 1] = mx;
}

// ---------------------------------------------------------------------------
// k3: pos_conf[cell] = sum_t fmax[t][cell]  (coalesced column reduce)
// ---------------------------------------------------------------------------
__global__ __launch_bounds__(128)
void k3_posconf(const float* __restrict__ ws, float* __restrict__ posconf) {
  const int cell = blockIdx.x * 128 + threadIdx.x;
  if (cell >= CELLS) return;
  const float* fm = ws + WS_FMAX;
  float s = 0.f;
  for (int t = 0; t < T_; ++t) s += fm[(size_t)t * CELLS + cell];
  posconf[cell] = s;
}

// ---------------------------------------------------------------------------
// k4: final combine (single block); deterministically reduces block partials.
// ---------------------------------------------------------------------------
__global__ __launch_bounds__(1024)
void k4_final(const float* __restrict__ finp, const int* __restrict__ target,
              const int* __restrict__ gmask, const int* __restrict__ eplen,
              const float* __restrict__ ws, const float* __restrict__ posconf,
              float* __restrict__ out_total) {
  const int j = threadIdx.x;
  __shared__ float sce[1024], sval[1024], slow[1024], skl[1024], sbi[1024];

  // per-cell terms
  float ce = 0.f, val = 0.f, low = 0.f;
  if (j < CELLS) {
    const bool mk = gmask[j] > 0;
    const float* fr = finp + (size_t)j * C_;
    float fi[C_];
    #pragma unroll
    for (int c = 0; c < C_; ++c) fi[c] = fr[c];
    float m = fi[0];
    #pragma unroll
    for (int c = 1; c < C_; ++c) m = fmaxf(m, fi[c]);
    float s = 0.f;
    #pragma unroll
    for (int c = 0; c < C_; ++c) s += __expf(fi[c] - m);
    const float lse = __logf(s);
    const int tg = target[j];
    if (mk) { ce = -(fi[tg] - m - lse); val = 1.f; }
    const float pc = posconf[j] * (1.f / (float)T_);
    if (mk && pc <= THRESH_) low = 1.f;
  }

  // deterministic reduction of main-pass block partials
  float klp = 0.f, bip = 0.f;
  for (int p = j; p < NBLK; p += 1024) {
    klp += ws[WS_PART_KL + p];
    bip += ws[WS_PART_BI + p];
  }

  sce[j] = ce; sval[j] = val; slow[j] = low; skl[j] = klp; sbi[j] = bip;
  __syncthreads();
  #pragma unroll
  for (int s2 = 512; s2 > 0; s2 >>= 1) {
    if (j < s2) {
      sce[j]  += sce[j + s2];
      sval[j] += sval[j + s2];
      slow[j] += slow[j + s2];
      skl[j]  += skl[j + s2];
      sbi[j]  += sbi[j + s2];
    }
    __syncthreads();
  }

  if (j == 0) {
    const float valid    = sval[0];
    const float inf_risk = slow[0] / (valid + EPS_);
    const float denom    = (float)T_ * valid + EPS_;
    const float risk = (skl[0] / denom) * (0.5f + 0.5f * (1.f - inf_risk));
    const float cons = sce[0] / (valid + EPS_);
    const float bi   = sbi[0] / denom;
    const float T_len = (float)eplen[0];
    const float grid_scale = (float)(H_ * W_) / 900.f;       // == 1.0
    float step = 0.1f * T_len * grid_scale;
    const float size_factor = sqrtf(grid_scale);
    const float log_factor  = logf(grid_scale + 1.f);
    step = step / (size_factor * (1.f + 0.1f * log_factor));
    const float gn = (grid_scale >= 0.1f && grid_scale <= 1.f)
                         ? 0.f : fabsf(grid_scale - 0.5f);
    // total = LAM_RISK*risk + LAM_AMB*amb(=0) + step + LAM_GN*gn + LAM_CONS*cons + LAM_BI*bi
    out_total[0] = risk + step + 0.1f * gn + cons + 0.5f * bi;
  }
}

extern "C" void kernel_launch(void* const* d_in, const int* in_sizes, int n_in,
                              void* d_out, int out_size, void* d_ws, size_t ws_size,
                              hipStream_t stream) {
  (void)in_sizes; (void)n_in; (void)out_size; (void)ws_size;
  const float* fpred = (const float*)d_in[0];
  const float* bpred = (const float*)d_in[1];
  // d_in[2] (state_predictions) and d_in[3] (observation_probs) are unused:
  // LAM_AMB == 0.0, so the ambiguity term never reaches any output.
  const float* finp   = (const float*)d_in[4];
  const float* pe     = (const float*)d_in[5];
  const float* Wp     = (const float*)d_in[6];
  const float* bpref  = (const float*)d_in[7];
  const int*   target = (const int*)d_in[8];
  const int*   gmask  = (const int*)d_in[9];
  const int*   eplen  = (const int*)d_in[10];

  float* ws       = (float*)d_ws;
  float* out      = (float*)d_out;
  float* out_conf = out + 1;
  float* out_mv   = out + 1 + (size_t)TOTAL;
  unsigned char* colors = (unsigned char*)(ws + WS_FMAX + (size_t)TOTAL);
  float* posconf  = ws + WS_POSCONF;

  k0_pref<<<1, 32, 0, stream>>>(pe, Wp, bpref, ws);
  k1_main<<<NBLK, 256, 0, stream>>>(fpred, bpred, gmask, ws, colors, out_conf);
  k2_move<<<NBLK, 256, 0, stream>>>(colors, out_mv);
  k3_posconf<<<(CELLS + 127) / 128, 128, 0, stream>>>(ws, posconf);
  k4_final<<<1, 1024, 0, stream>>>(finp, target, gmask, eplen, ws, posconf, out);
}